// RoleBasedProjection_10075993277119
// MI455X (gfx1250) — compile-verified
//
#include <hip/hip_runtime.h>

typedef __attribute__((ext_vector_type(2))) float v2f;
typedef __attribute__((ext_vector_type(8))) float v8f;

#define BM 128          // rows per workgroup macro-tile
#define BN 128          // cols per workgroup macro-tile (8 waves x 16)
#define KC 16           // K chunk staged in LDS
#define LDA 20          // padded LDS row stride in floats (bank-conflict free, 16B aligned)

__global__ __launch_bounds__(256, 2)
void role_proj_kernel(const float* __restrict__ X,
                      const int* __restrict__ tmask,
                      const int* __restrict__ imask,
                      const float* __restrict__ Wt, const float* __restrict__ bt,
                      const float* __restrict__ Wi, const float* __restrict__ bi,
                      const float* __restrict__ Wo, const float* __restrict__ bo,
                      float* __restrict__ Out,
                      int C, int H, int T)
{
    __shared__ float Alds[2][BM * LDA];

    const int tid  = threadIdx.x;
    const int lane = tid & 31;
    const int wv   = tid >> 5;      // wave id 0..7 -> 16-col N strip
    const int nn   = lane & 15;     // N index within tile (B/C/D) and M index (A)
    const int kh   = lane >> 4;     // K-half selector (0 or 1)

    const int rowBase = blockIdx.x * BM;
    const int colBase = blockIdx.y * BN + wv * 16;

    // Bug-faithful role selection: mask index = row / T. BM divides T, so uniform per block.
    const int g  = rowBase / T;
    const int tm = (tmask[g] != 0);
    const int im = (imask[g] != 0) && !tm;
    const float* __restrict__ W = tm ? Wt : (im ? Wi : Wo);
    const float* __restrict__ B = tm ? bt : (im ? bi : bo);

    // Per-thread staging geometry: 2 async b128 ops cover the 128x16 chunk.
    const int r0  = tid >> 2;            // 0..63   (pass 0 rows)
    const int r1  = (tid + 256) >> 2;    // 64..127 (pass 1 rows)
    const int c4  = (tid & 3) * 4;       // 0,4,8,12

    // Async DMA of one A chunk (BM x KC floats) into LDS buffer `buf`.
    auto issue_chunk = [&](int buf, int k0) {
        {
            const float* gp = X + (size_t)(rowBase + r0) * C + k0 + c4;
            unsigned ld = (unsigned)(uintptr_t)&Alds[buf][r0 * LDA + c4];
            asm volatile("global_load_async_to_lds_b128 %0, %1, off"
                         :: "v"(ld), "v"(gp) : "memory");
        }
        {
            const float* gp = X + (size_t)(rowBase + r1) * C + k0 + c4;
            unsigned ld = (unsigned)(uintptr_t)&Alds[buf][r1 * LDA + c4];
            asm volatile("global_load_async_to_lds_b128 %0, %1, off"
                         :: "v"(ld), "v"(gp) : "memory");
        }
    };

    v8f acc[8];
    #pragma unroll
    for (int i = 0; i < 8; ++i) acc[i] = (v8f){0.f,0.f,0.f,0.f,0.f,0.f,0.f,0.f};

    const int NC = C / KC;               // 16 chunks
    issue_chunk(0, 0);

    for (int kc = 0; kc < NC; ++kc) {
        const int cur = kc & 1;
        const int k0  = kc * KC;

        if (kc + 1 < NC) {
            issue_chunk(1 - cur, k0 + KC);          // prefetch next chunk
            asm volatile("s_wait_asynccnt 0x2" ::: "memory");  // oldest chunk landed
        } else {
            asm volatile("s_wait_asynccnt 0x0" ::: "memory");
        }
        __syncthreads();                             // chunk kc visible to all waves

        #pragma unroll
        for (int ks = 0; ks < KC; ks += 4) {
            const int kk2 = ks + 2 * kh;  // this lane-half's K pair within the chunk
            // B fragment (4x16): VGPR0 holds K=kk2, VGPR1 holds K=kk2+1, N = nn
            v2f b;
            b.x = W[(size_t)(k0 + kk2)     * H + colBase + nn];
            b.y = W[(size_t)(k0 + kk2 + 1) * H + colBase + nn];
            #pragma unroll
            for (int mt = 0; mt < 8; ++mt) {
                // A fragment (16x4): M = nn, K pair at kk2 (8B-aligned ds_load_b64)
                v2f a = *(const v2f*)(&Alds[cur][(mt * 16 + nn) * LDA + kk2]);
                acc[mt] = __builtin_amdgcn_wmma_f32_16x16x4_f32(
                    false, a, false, b, (short)0, acc[mt], false, false);
            }
        }
        __syncthreads();                             // safe to overwrite buf[cur] later
    }

    // Epilogue: bias add + coalesced stores (C/D layout: row = 8*kh + r, col = nn)
    const float bb = B[colBase + nn];
    #pragma unroll
    for (int mt = 0; mt < 8; ++mt) {
        #pragma unroll
        for (int r = 0; r < 8; ++r) {
            size_t row = (size_t)rowBase + mt * 16 + 8 * kh + r;
            Out[row * H + colBase + nn] = acc[mt][r] + bb;
        }
    }
}

extern "C" void kernel_launch(void* const* d_in, const int* in_sizes, int n_in,
                              void* d_out, int out_size, void* d_ws, size_t ws_size,
                              hipStream_t stream) {
    const float* X  = (const float*)d_in[0];
    const int*   tmv = (const int*)d_in[1];
    const int*   imv = (const int*)d_in[2];
    const float* Wt = (const float*)d_in[3];
    const float* bt = (const float*)d_in[4];
    const float* Wi = (const float*)d_in[5];
    const float* bi = (const float*)d_in[6];
    const float* Wo = (const float*)d_in[7];
    const float* bo = (const float*)d_in[8];
    float* Out = (float*)d_out;

    const int nvars = in_sizes[1];
    const int H = in_sizes[4];              // hidden = 512
    const int C = in_sizes[3] / H;          // channels = 256
    const int M = in_sizes[0] / C;          // T * n_vars = 131072
    const int T = M / nvars;                // 2048

    dim3 grid(M / BM, H / BN);
    role_proj_kernel<<<grid, 256, 0, stream>>>(X, tmv, imv, Wt, bt, Wi, bi, Wo, bo,
                                               Out, C, H, T);
}